// Model_3315714752612
// MI455X (gfx1250) — compile-verified
//
#include <hip/hip_runtime.h>
#include <math.h>

typedef __attribute__((ext_vector_type(2)))  float        v2f;
typedef __attribute__((ext_vector_type(8)))  float        v8f;
typedef __attribute__((ext_vector_type(4)))  unsigned int v4u;
typedef __attribute__((ext_vector_type(8)))  int          v8i;
typedef __attribute__((ext_vector_type(4)))  int          v4i;

#define L_SEQ   2048
#define H_DIM   512
#define N_ST    32
#define IN_DIM  4
#define OUT_DIM 3
#define N_LAYER 4
// TDM pads 4 dwords after every 256 dwords -> 512-dword row becomes 520 dwords in LDS
#define TROW    520

__device__ __forceinline__ float gelu_f(float x) {
    float x3 = x * x * x;
    return 0.5f * x * (1.f + tanhf(0.7978845608028654f * (x + 0.044715f * x3)));
}
__device__ __forceinline__ float sigmoid_f(float x) {
    return 1.f / (1.f + __expf(-x));
}

// ---- Tensor Data Mover: load a 16x512 f32 tile (row stride 512) into LDS with padding.
// D# packing per CDNA5 ISA.
__device__ __forceinline__ void tdm_load_tile16x512(const float* gptr, unsigned lds_off) {
    unsigned long long ga = (unsigned long long)(size_t)gptr;
    v4u g0;
    g0.x = 1u;                                   // count=1, user descriptor
    g0.y = lds_off;                              // lds_addr (bytes)
    g0.z = (unsigned)(ga & 0xFFFFFFFFu);         // global_addr[31:0]
    g0.w = (unsigned)((ga >> 32) & 0x1FFFFFFu)   // global_addr[56:32]
         | (2u << 30);                           // type=2 ("image")
    v8i g1;
    g1[0] = (2 << 16)       // data_size = 4 bytes
          | (1 << 20)       // pad_enable
          | (7 << 22)       // pad_interval: every 256 dwords
          | (3 << 25);      // pad_amount: 4 dwords
    g1[1] = (512 << 16);    // tensor_dim0[15:0] = 512 (barrier addr = 0)
    g1[2] = (512 << 16);    // tensor_dim0 hi = 0 | tensor_dim1[15:0] = 512
    g1[3] = (512 << 16);    // tensor_dim1 hi = 0 | tile_dim0 = 512
    g1[4] = 16;             // tile_dim1 = 16, tile_dim2 = 0
    g1[5] = 512;            // tensor_dim0_stride lo32 = 512
    g1[6] = 0;              // dim0_stride hi | dim1_stride lo16
    g1[7] = 4;              // tensor_dim1_stride = 512*512 (hi bits)
    v4i g2 = {0, 0, 0, 0};
    v4i g3 = {0, 0, 0, 0};
#if __clang_major__ >= 23
    v8i g4 = {0, 0, 0, 0, 0, 0, 0, 0};
    __builtin_amdgcn_tensor_load_to_lds(g0, g1, g2, g3, g4, 0);
#else
    __builtin_amdgcn_tensor_load_to_lds(g0, g1, g2, g3, 0);
#endif
}

// ---------------- Encoder: h[L,H] = x[L,4] @ enc_w.T + enc_b  (K=4 == one WMMA) ----
__global__ void encoder_kernel(const float* __restrict__ x,
                               const float* __restrict__ w,   // [H, 4]
                               const float* __restrict__ b,   // [H]
                               float* __restrict__ h) {
    int wid  = threadIdx.x >> 5;
    int lane = threadIdx.x & 31;
    int tile = blockIdx.x * 4 + wid;        // 128 x 32 = 4096 tiles
    int tm = tile >> 5;
    int tn = tile & 31;
    int mn = lane & 15;
    int khalf = (lane < 16) ? 0 : 2;

    v2f a  = *reinterpret_cast<const v2f*>(x + (tm * 16 + mn) * IN_DIM + khalf);
    v2f bf = *reinterpret_cast<const v2f*>(w + (tn * 16 + mn) * IN_DIM + khalf);
    v8f acc = {};
    acc = __builtin_amdgcn_wmma_f32_16x16x4_f32(false, a, false, bf, (short)0, acc,
                                                false, false);
    float bias = b[tn * 16 + mn];
    int rbase = tm * 16 + ((lane < 16) ? 0 : 8);
#pragma unroll
    for (int r = 0; r < 8; ++r)
        h[(rbase + r) * H_DIM + tn * 16 + mn] = acc[r] + bias;
}

// ---------------- LayerNorm: u = LN(h)*nw + nb, one block (512 thr) per row --------
__global__ void layernorm_kernel(const float* __restrict__ h,
                                 const float* __restrict__ nw,
                                 const float* __restrict__ nb,
                                 float* __restrict__ u) {
    __shared__ float r1[16], r2[16];
    int row = blockIdx.x, tid = threadIdx.x;
    float v = h[row * H_DIM + tid];
    float s1 = v, s2 = v * v;
#pragma unroll
    for (int off = 16; off > 0; off >>= 1) {
        s1 += __shfl_xor(s1, off, 32);
        s2 += __shfl_xor(s2, off, 32);
    }
    int wid = tid >> 5, lane = tid & 31;
    if (lane == 0) { r1[wid] = s1; r2[wid] = s2; }
    __syncthreads();
    float a = r1[lane & 15], b = r2[lane & 15];
#pragma unroll
    for (int off = 8; off > 0; off >>= 1) {
        a += __shfl_xor(a, off, 32);
        b += __shfl_xor(b, off, 32);
    }
    float mean = a * (1.f / H_DIM);
    float var  = b * (1.f / H_DIM) - mean * mean;
    float inv  = rsqrtf(var + 1e-5f);
    u[row * H_DIM + tid] = (v - mean) * inv * nw[tid] + nb[tid];
}

// ---------------- S4 scan: one wave32 per channel; lane n = state n (N==32) -------
__global__ void s4_scan_kernel(const float* __restrict__ u,
                               const float* __restrict__ lre, const float* __restrict__ lim,
                               const float* __restrict__ bre, const float* __restrict__ bim,
                               const float* __restrict__ cre, const float* __restrict__ cim,
                               const float* __restrict__ dvec, const float* __restrict__ lstep,
                               float* __restrict__ y) {
    int wid = threadIdx.x >> 5, lane = threadIdx.x & 31;
    int ch  = blockIdx.x * 8 + wid;
    int idx = ch * N_ST + lane;

    float step = __expf(lstep[ch]);
    float lr = lre[idx], li = lim[idx];
    float hs = 0.5f * step;
    float dr = 1.f - hs * lr, di = -hs * li;
    float den = 1.f / (dr * dr + di * di);
    float nr = 1.f + hs * lr, ni = hs * li;
    float ar = (nr * dr + ni * di) * den;
    float ai = (ni * dr - nr * di) * den;
    float br0 = bre[idx], bi0 = bim[idx];
    float tr = (dr * br0 + di * bi0) * den * step;
    float ti = (dr * bi0 - di * br0) * den * step;
    float cr = cre[idx], ci = cim[idx];
    float dch = dvec[ch];

    float sre = 0.f, sim = 0.f;
    for (int t0 = 0; t0 < L_SEQ; t0 += 32) {
        float uv = u[(t0 + lane) * H_DIM + ch];
        float yo = 0.f;
#pragma unroll
        for (int k = 0; k < 32; ++k) {
            float uk = __shfl(uv, k, 32);
            float n_re = ar * sre - ai * sim + tr * uk;
            float n_im = ar * sim + ai * sre + ti * uk;
            sre = n_re; sim = n_im;
            float p = cr * sre - ci * sim;
#pragma unroll
            for (int off = 16; off > 0; off >>= 1) p += __shfl_xor(p, off, 32);
            float yv = 2.f * p + dch * uk;
            if (lane == k) yo = gelu_f(yv);
        }
        y[(t0 + lane) * H_DIM + ch] = yo;
    }
}

// ---------------- GLU: h += (y@w1.T + b1) * sigmoid(y@w2.T + b2) -------------------
// Block = 4 waves = 4 M-tiles sharing one N-tile; w1/w2 N-tiles staged in LDS by TDM.
__global__ void glu_matmul_kernel(const float* __restrict__ y,
                                  const float* __restrict__ w1, const float* __restrict__ b1v,
                                  const float* __restrict__ w2, const float* __restrict__ b2v,
                                  float* __restrict__ h) {
    __shared__ float lw[2 * 16 * TROW];     // two padded 16x512 tiles (65 KB)
    int wid = threadIdx.x >> 5, lane = threadIdx.x & 31;
    int tn = blockIdx.x & 31;               // N tile (shared by block)
    int tm = (blockIdx.x >> 5) * 4 + wid;   // M tile (per wave)
    int mn = lane & 15;
    int khalf = (lane < 16) ? 0 : 2;

    unsigned lds0 = (unsigned)(size_t)(void*)&lw[0];
    if (wid == 0)      tdm_load_tile16x512(w1 + tn * 16 * H_DIM, lds0);
    else if (wid == 1) tdm_load_tile16x512(w2 + tn * 16 * H_DIM,
                                           lds0 + 16 * TROW * sizeof(float));
    __builtin_amdgcn_s_wait_tensorcnt(0);
    __syncthreads();

    const float* arow = y  + (tm * 16 + mn) * H_DIM + khalf;
    const float* l1 = lw + mn * TROW + khalf;
    const float* l2 = lw + (16 + mn) * TROW + khalf;

    v8f acc1 = {}, acc2 = {};
    // k = 0..255 (before mid-row pad)
#pragma unroll 4
    for (int kb = 0; kb < 256; kb += 4) {
        __builtin_prefetch(arow + kb + 64, 0, 3);
        v2f a = *reinterpret_cast<const v2f*>(arow + kb);
        v2f p = *reinterpret_cast<const v2f*>(l1 + kb);
        v2f q = *reinterpret_cast<const v2f*>(l2 + kb);
        acc1 = __builtin_amdgcn_wmma_f32_16x16x4_f32(false, a, false, p, (short)0, acc1,
                                                     false, false);
        acc2 = __builtin_amdgcn_wmma_f32_16x16x4_f32(false, a, false, q, (short)0, acc2,
                                                     false, false);
    }
    // k = 256..511 (LDS data shifted +4 dwords past the pad)
#pragma unroll 4
    for (int kb = 0; kb < 256; kb += 4) {
        v2f a = *reinterpret_cast<const v2f*>(arow + 256 + kb);
        v2f p = *reinterpret_cast<const v2f*>(l1 + 260 + kb);
        v2f q = *reinterpret_cast<const v2f*>(l2 + 260 + kb);
        acc1 = __builtin_amdgcn_wmma_f32_16x16x4_f32(false, a, false, p, (short)0, acc1,
                                                     false, false);
        acc2 = __builtin_amdgcn_wmma_f32_16x16x4_f32(false, a, false, q, (short)0, acc2,
                                                     false, false);
    }

    int gcol = tn * 16 + mn;
    float bb1 = b1v[gcol], bb2 = b2v[gcol];
    int rbase = tm * 16 + ((lane < 16) ? 0 : 8);
#pragma unroll
    for (int r = 0; r < 8; ++r) {
        float v1 = acc1[r] + bb1;
        float v2 = acc2[r] + bb2;
        h[(rbase + r) * H_DIM + gcol] += v1 * sigmoid_f(v2);
    }
}

// ---------------- Decoder: out[L,3] = h @ dec_w.T + dec_b --------------------------
__global__ void decoder_kernel(const float* __restrict__ h, const float* __restrict__ w,
                               const float* __restrict__ b, float* __restrict__ out) {
    int idx = blockIdx.x * blockDim.x + threadIdx.x;
    if (idx >= L_SEQ * OUT_DIM) return;
    int l = idx / OUT_DIM, o = idx % OUT_DIM;
    const float* hr = h + l * H_DIM;
    const float* wr = w + o * H_DIM;
    float s = b[o];
#pragma unroll 8
    for (int k = 0; k < H_DIM; ++k) s += hr[k] * wr[k];
    out[idx] = s;
}

extern "C" void kernel_launch(void* const* d_in, const int* in_sizes, int n_in,
                              void* d_out, int out_size, void* d_ws, size_t ws_size,
                              hipStream_t stream) {
    const float* x       = (const float*)d_in[0];
    const float* enc_w   = (const float*)d_in[1];
    const float* enc_b   = (const float*)d_in[2];
    const float* dec_w   = (const float*)d_in[3];
    const float* dec_b   = (const float*)d_in[4];
    const float* norm_w  = (const float*)d_in[5];
    const float* norm_b  = (const float*)d_in[6];
    const float* lam_re  = (const float*)d_in[7];
    const float* lam_im  = (const float*)d_in[8];
    const float* b_re    = (const float*)d_in[9];
    const float* b_im    = (const float*)d_in[10];
    const float* c_re    = (const float*)d_in[11];
    const float* c_im    = (const float*)d_in[12];
    const float* dd      = (const float*)d_in[13];
    const float* lstep   = (const float*)d_in[14];
    const float* out_w   = (const float*)d_in[15];
    const float* out_b   = (const float*)d_in[16];
    const float* out2_w  = (const float*)d_in[17];
    const float* out2_b  = (const float*)d_in[18];

    float* u  = (float*)d_ws;               // [L,H] layernorm output
    float* yb = u  + L_SEQ * H_DIM;         // [L,H] scan+GELU output
    float* h  = yb + L_SEQ * H_DIM;         // [L,H] residual stream

    encoder_kernel<<<1024, 128, 0, stream>>>(x, enc_w, enc_b, h);
    for (int i = 0; i < N_LAYER; ++i) {
        int hn = i * H_DIM * N_ST;
        layernorm_kernel<<<L_SEQ, H_DIM, 0, stream>>>(h, norm_w + i * H_DIM,
                                                      norm_b + i * H_DIM, u);
        s4_scan_kernel<<<H_DIM / 8, 256, 0, stream>>>(u, lam_re + hn, lam_im + hn,
                                                      b_re + hn, b_im + hn,
                                                      c_re + hn, c_im + hn,
                                                      dd + i * H_DIM, lstep + i * H_DIM, yb);
        glu_matmul_kernel<<<1024, 128, 0, stream>>>(yb, out_w + i * H_DIM * H_DIM,
                                                    out_b + i * H_DIM,
                                                    out2_w + i * H_DIM * H_DIM,
                                                    out2_b + i * H_DIM, h);
    }
    decoder_kernel<<<(L_SEQ * OUT_DIM + 255) / 256, 256, 0, stream>>>(h, dec_w, dec_b,
                                                                      (float*)d_out);
}